// Grid_nd_sample_17961553232033
// MI455X (gfx1250) — compile-verified
//
#include <hip/hip_runtime.h>

// Grid-sample (bilinear) for in_tensor (B,H,W,C) f32, indices (B,P,2) f32.
// Memory-bound gather; the 4-neighbor blend runs on the CDNA5 matrix pipe via
// V_WMMA_F32_16X16X4_F32. A = 16x4 replicated weight rows; B = 4 neighbors x
// 16 channels. Two chunks per step so D-row replication lets the full wave
// store 128B contiguously with no exec-mask branching.

typedef __attribute__((ext_vector_type(2))) float v2f;
typedef __attribute__((ext_vector_type(8))) float v8f;

#define B_  16
#define H_  128
#define W_  128
#define C_  256
#define P_  8192

__global__ __launch_bounds__(256)
void grid_sample_wmma_f32(const float* __restrict__ in,
                          const float* __restrict__ idx,
                          float* __restrict__ out) {
    const int lane = threadIdx.x & 31;
    const int wave = threadIdx.x >> 5;
    const int pt   = blockIdx.x * 8 + wave;      // point index in [0, B*P)
    const int b    = pt >> 13;                   // pt / P_   (P_ = 8192)

    // per-point scalars (wave-uniform)
    const float y = idx[pt * 2 + 0];             // indexes H (reference dim 0)
    const float x = idx[pt * 2 + 1];             // indexes W (reference dim 1)
    const float fy = floorf(y), cy = ceilf(y);
    const float fx = floorf(x), cx = ceilf(x);
    const float mx = y - fy;                     // mu_x in reference
    const float my = x - fx;                     // mu_y in reference
    const int ify = (int)fy, icy = (int)cy, ifx = (int)fx, icx = (int)cx;

    // out = p1*(1-mx)(1-my) + p2*mx(1-my) + p3*(1-mx)my + p4*mx*my
    const float w0 = (1.0f - mx) * (1.0f - my);  // (fy, fx)
    const float w1 = mx * (1.0f - my);           // (cy, fx)
    const float w2 = (1.0f - mx) * my;           // (fy, cx)
    const float w3 = mx * my;                    // (cy, cx)

    const size_t base = (size_t)b * (size_t)(H_ * W_ * C_);
    const float* r0 = in + base + ((size_t)(ify * W_ + ifx) * C_);
    const float* r1 = in + base + ((size_t)(icy * W_ + ifx) * C_);
    const float* r2 = in + base + ((size_t)(ify * W_ + icx) * C_);
    const float* r3 = in + base + ((size_t)(icy * W_ + icx) * C_);

    // WMMA f32 16x16x4 operand layout (ISA 7.12.2):
    //   A (16x4): lanes 0-15 -> K=0,1 in vgpr0,1 ; lanes 16-31 -> K=2,3
    //   B (4x16): lane(half,n) -> rows K=2*half, 2*half+1 at column n
    const int half = lane >> 4;
    const int n    = lane & 15;

    const float* rA = half ? r2 : r0;            // K = 2*half
    const float* rB = half ? r3 : r1;            // K = 2*half + 1
    v2f a;
    a.x = half ? w2 : w0;
    a.y = half ? w3 : w1;

    float* o = out + (size_t)pt * C_;

    // Gather all 4 KB of neighbor-row data first: 32 coalesced b32 loads per
    // lane, clause-able, max loads in flight (memory-level parallelism).
    float d0[8], d1[8], d2[8], d3[8];
#pragma unroll
    for (int i = 0; i < 8; ++i) {
        const int cb = 32 * i + n;
        d0[i] = rA[cb];                          // chunk 2i,   row 2*half
        d1[i] = rB[cb];                          // chunk 2i,   row 2*half+1
        d2[i] = rA[cb + 16];                     // chunk 2i+1, row 2*half
        d3[i] = rB[cb + 16];                     // chunk 2i+1, row 2*half+1
    }

    // Blend on the matrix pipe: 2 WMMAs per step (chunks 2i, 2i+1).
    // D rows are replicated, so lanes 0-15 carry chunk 2i in acc0[0] and
    // lanes 16-31 carry chunk 2i+1 in acc1[0] -> one contiguous 128B store:
    // offset (2i+half)*16 + n == 32*i + lane.
#pragma unroll
    for (int i = 0; i < 8; ++i) {
        v2f b0; b0.x = d0[i]; b0.y = d1[i];
        v2f b1; b1.x = d2[i]; b1.y = d3[i];
        v8f acc0 = {}, acc1 = {};
        acc0 = __builtin_amdgcn_wmma_f32_16x16x4_f32(
            false, a, false, b0, (short)0, acc0, false, false);
        acc1 = __builtin_amdgcn_wmma_f32_16x16x4_f32(
            false, a, false, b1, (short)0, acc1, false, false);
        o[32 * i + lane] = half ? acc1[0] : acc0[0];
    }
}

extern "C" void kernel_launch(void* const* d_in, const int* in_sizes, int n_in,
                              void* d_out, int out_size, void* d_ws, size_t ws_size,
                              hipStream_t stream) {
    const float* in_tensor = (const float*)d_in[0];
    const float* indices   = (const float*)d_in[1];
    float* out = (float*)d_out;

    // 8 waves (one point each) per 256-thread block
    const int total_points = B_ * P_;            // 131072
    dim3 grid(total_points / 8);                 // 16384 blocks
    dim3 block(256);
    grid_sample_wmma_f32<<<grid, block, 0, stream>>>(in_tensor, indices, out);
}